// VFGNN_58136677319060
// MI455X (gfx1250) — compile-verified
//
#include <hip/hip_runtime.h>
#include <hip/hip_bf16.h>

// -------------------------------------------------------------------------
// VFGNN forward on MI455X (gfx1250, wave32).
// GEMMs use V_WMMA_F32_16X16X4_F32 (exact fp32 semantics; the graph
// scatter/gather over L2 dominates runtime, so fp32 WMMA costs nothing and
// avoids bf16 rounding across 4 chained layers).
//
// Per block: 8 waves = 128 rows x 16 cols of output. The 16-column weight
// slab(s) are staged once into LDS (padded column-major, conflict-free
// ds_load_b64 fragment reads). K and H are template constants so the fully
// unrolled inner loop uses immediate-offset addressing only.
//
// f32 WMMA fragment layout (CDNA5 ISA 7.12.2):
//  A (16x4, v2f): lane L: row M = L&15, K pair = {2*(L>>4), 2*(L>>4)+1}
//  B (4x16, v2f): lane L: col N = L&15, K pair = {2*(L>>4), 2*(L>>4)+1}
//  C/D (16x16, v8f): VGPR i: row M = i + 8*(L>>4), col N = L&15
// -------------------------------------------------------------------------

typedef float v2f __attribute__((ext_vector_type(2)));
typedef float v8f __attribute__((ext_vector_type(8)));

__global__ void zero_f32(float* __restrict__ p, long n) {
    long t = (long)blockIdx.x * blockDim.x + threadIdx.x;
    if (t < n) p[t] = 0.0f;
}

__global__ void degree_count(const int* __restrict__ dst, float* __restrict__ deg, long E) {
    long t = (long)blockIdx.x * blockDim.x + threadIdx.x;
    if (t < E) atomicAdd(&deg[dst[t]], 1.0f);
}

__global__ void invert_deg(float* __restrict__ deg, long n) {
    long t = (long)blockIdx.x * blockDim.x + threadIdx.x;
    if (t < n) {
        float d = deg[t];
        deg[t] = (d > 0.0f) ? (1.0f / d) : 0.0f;
    }
}

// One thread = one (edge, 4-feature chunk); F=128 -> 32 chunks/edge.
// Accumulates h[src]*invdeg[dst] so the output buffer holds the MEAN directly
// (invdeg[dst] is constant per destination: sum(h*c) == c*sum(h)).
__global__ void scatter_mean(const float* __restrict__ h, const int* __restrict__ src,
                             const int* __restrict__ dst, const float* __restrict__ invdeg,
                             float* __restrict__ mean, long nwork) {
    long t = (long)blockIdx.x * blockDim.x + threadIdx.x;
    if (t >= nwork) return;
    long e = t >> 5;
    int  c = (int)(t & 31) * 4;
    long s = src[e];
    long d = dst[e];
    const float w = invdeg[d];
    const float* hp = h + s * 128 + c;
    float*       ap = mean + d * 128 + c;
    atomicAdd(ap + 0, hp[0] * w);
    atomicAdd(ap + 1, hp[1] * w);
    atomicAdd(ap + 2, hp[2] * w);
    atomicAdd(ap + 3, hp[3] * w);
}

// Out[m, outoff+n] = act( A[m,:]@Wself[:,n] (+ Mean[m,:]@Wneigh[:,n]) + bias[n] )
// A,Mean: [Nrows x K] row-major.  Wself,Wneigh: [K x H] row-major.
template <int K, int H, bool NEIGH, bool RELU>
__global__ __launch_bounds__(256) void sage_gemm(
        const float* __restrict__ A, const float* __restrict__ Mean,
        const float* __restrict__ Wself, const float* __restrict__ Wneigh,
        const float* __restrict__ bias, float* __restrict__ Out,
        int Nrows, int ldo, int outoff) {
    constexpr int KP = K + 2;                    // pad -> bank stride 2 dwords
    __shared__ float Bs[16 * KP];
    __shared__ float Bn[NEIGH ? 16 * KP : 1];

    const int wave = threadIdx.x >> 5;
    const int lane = threadIdx.x & 31;
    const int half = lane >> 4;
    const int lid  = lane & 15;

    const int rowbase = (blockIdx.x * 8 + wave) * 16;
    const int colbase = blockIdx.y * 16;

    // Cooperative stage: i -> (k = i/16, c = i%16). 16 consecutive threads read
    // 16 consecutive columns of weight row k (coalesced); LDS writes hit
    // distinct banks (stride KP dwords -> bank stride 2).
    for (int i = threadIdx.x; i < 16 * K; i += 256) {
        const int k = i >> 4;
        const int c = i & 15;
        const int n = colbase + c;
        const int nc = (n < H) ? n : (H - 1);    // clamp: cols are independent
        Bs[c * KP + k] = Wself[k * H + nc];
        if (NEIGH) Bn[c * KP + k] = Wneigh[k * H + nc];
    }
    __syncthreads();

    if (rowbase >= Nrows) return;                // after barrier; wave-uniform

    const int n  = colbase + lid;
    const int nc = (n < H) ? n : (H - 1);
    const int m  = rowbase + lid;
    const int mc = (m < Nrows) ? m : (Nrows - 1);
    const int kk = half * 2;                     // lane-half -> K pair {kk,kk+1}

    const float* Arow  = A + (long)mc * K + kk;
    const float* Grow  = NEIGH ? (Mean + (long)mc * K + kk) : nullptr;
    const float* bsrow = &Bs[lid * KP + kk];
    const float* bnrow = NEIGH ? &Bn[lid * KP + kk] : nullptr;

    v8f cs = {};
    v8f cn = {};

    #pragma unroll
    for (int k = 0; k < K; k += 4) {
        v2f a = *(const v2f*)(Arow + k);
        v2f b = *(const v2f*)(bsrow + k);
        cs = __builtin_amdgcn_wmma_f32_16x16x4_f32(false, a, false, b,
                                                   (short)0, cs, false, false);
        if (NEIGH) {
            v2f g  = *(const v2f*)(Grow + k);
            v2f bb = *(const v2f*)(bnrow + k);
            cn = __builtin_amdgcn_wmma_f32_16x16x4_f32(false, g, false, bb,
                                                       (short)0, cn, false, false);
        }
    }

    const float bv = bias[nc];
    #pragma unroll
    for (int i = 0; i < 8; ++i) {
        float v = cs[i] + (NEIGH ? cn[i] : 0.0f) + bv;
        if (RELU) v = (v > 0.0f) ? v : 0.0f;
        const int row = rowbase + i + 8 * half;
        if (n < H && row < Nrows)
            Out[(long)row * ldo + outoff + n] = v;
    }
}

static inline long cdivl(long a, long b) { return (a + b - 1) / b; }

extern "C" void kernel_launch(void* const* d_in, const int* in_sizes, int n_in,
                              void* d_out, int out_size, void* d_ws, size_t ws_size,
                              hipStream_t stream) {
    const float* x0   = (const float*)d_in[0];
    const float* x1   = (const float*)d_in[1];
    const int*   src0 = (const int*)d_in[2];
    const int*   dst0 = (const int*)d_in[3];
    const int*   src1 = (const int*)d_in[4];
    const int*   dst1 = (const int*)d_in[5];
    const float* ws0a = (const float*)d_in[6];
    const float* wn0a = (const float*)d_in[7];
    const float* b0a  = (const float*)d_in[8];
    const float* ws1a = (const float*)d_in[9];
    const float* wn1a = (const float*)d_in[10];
    const float* b1a  = (const float*)d_in[11];
    const float* ws0b = (const float*)d_in[12];
    const float* wn0b = (const float*)d_in[13];
    const float* b0b  = (const float*)d_in[14];
    const float* ws1b = (const float*)d_in[15];
    const float* wn1b = (const float*)d_in[16];
    const float* b1b  = (const float*)d_in[17];
    const float* wf1  = (const float*)d_in[18];
    const float* bf1  = (const float*)d_in[19];
    const float* wf2  = (const float*)d_in[20];
    const float* bf2  = (const float*)d_in[21];
    float* out = (float*)d_out;

    const int  D  = 128, H = 128, FH = 256, C = 40;
    const int  N  = in_sizes[0] / D;       // 100000
    const long E0 = in_sizes[2];
    const long E1 = in_sizes[4];

    // scratch layout
    float* ws   = (float*)d_ws;
    float* mean = ws;  ws += (long)N * 128;
    float* deg0 = ws;  ws += N;             // deg0,deg1 contiguous
    float* deg1 = ws;  ws += N;
    float* ha1  = ws;  ws += (long)N * 128;
    float* hb1  = ws;  ws += (long)N * 128;
    float* hcat = ws;  ws += (long)N * 256;
    float* hf   = ws;  ws += (long)N * 256;

    const int TB = 256;
    const long aggN = (long)N * 128;
    const int rowBlocks = (int)cdivl(cdivl(N, 16), 8);   // 8 waves/block
    const dim3 gH  (rowBlocks, H  / 16);
    const dim3 gFH (rowBlocks, FH / 16);
    const dim3 gC  (rowBlocks, (C + 15) / 16);

    // --- degrees -> reciprocal (both graphs) ---
    zero_f32<<<cdivl(2L * N, TB), TB, 0, stream>>>(deg0, 2L * N);
    degree_count<<<cdivl(E0, TB), TB, 0, stream>>>(dst0, deg0, E0);
    degree_count<<<cdivl(E1, TB), TB, 0, stream>>>(dst1, deg1, E1);
    invert_deg<<<cdivl(2L * N, TB), TB, 0, stream>>>(deg0, 2L * N);

    // --- party a, layer 1: x0 -> ha1 ---
    zero_f32<<<cdivl(aggN, TB), TB, 0, stream>>>(mean, aggN);
    scatter_mean<<<cdivl(E0 * 32, TB), TB, 0, stream>>>(x0, src0, dst0, deg0, mean, E0 * 32);
    sage_gemm<128, 128, true, true><<<gH, TB, 0, stream>>>(x0, mean, ws0a, wn0a, b0a,
                                                           ha1, N, H, 0);
    // --- party a, layer 2: ha1 -> hcat[:, 0:128] ---
    zero_f32<<<cdivl(aggN, TB), TB, 0, stream>>>(mean, aggN);
    scatter_mean<<<cdivl(E0 * 32, TB), TB, 0, stream>>>(ha1, src0, dst0, deg0, mean, E0 * 32);
    sage_gemm<128, 128, true, true><<<gH, TB, 0, stream>>>(ha1, mean, ws1a, wn1a, b1a,
                                                           hcat, N, 2 * H, 0);
    // --- party b, layer 1: x1 -> hb1 ---
    zero_f32<<<cdivl(aggN, TB), TB, 0, stream>>>(mean, aggN);
    scatter_mean<<<cdivl(E1 * 32, TB), TB, 0, stream>>>(x1, src1, dst1, deg1, mean, E1 * 32);
    sage_gemm<128, 128, true, true><<<gH, TB, 0, stream>>>(x1, mean, ws0b, wn0b, b0b,
                                                           hb1, N, H, 0);
    // --- party b, layer 2: hb1 -> hcat[:, 128:256] ---
    zero_f32<<<cdivl(aggN, TB), TB, 0, stream>>>(mean, aggN);
    scatter_mean<<<cdivl(E1 * 32, TB), TB, 0, stream>>>(hb1, src1, dst1, deg1, mean, E1 * 32);
    sage_gemm<128, 128, true, true><<<gH, TB, 0, stream>>>(hb1, mean, ws1b, wn1b, b1b,
                                                           hcat, N, 2 * H, H);
    // --- fuse: relu(hcat @ wf1 + bf1) -> hf ---
    sage_gemm<256, 256, false, true><<<gFH, TB, 0, stream>>>(hcat, nullptr, wf1, nullptr,
                                                             bf1, hf, N, FH, 0);
    // --- classifier: hf @ wf2 + bf2 -> out [N x 40] ---
    sage_gemm<256, 40, false, false><<<gC, TB, 0, stream>>>(hf, nullptr, wf2, nullptr,
                                                            bf2, out, N, C, 0);
}